// Interp2xBoundary2d_87041807221175
// MI455X (gfx1250) — compile-verified
//
#include <hip/hip_runtime.h>
#include <hip/hip_bf16.h>

// 2x bilinear upsample (half-pixel) + boundary mask for (4,8,512,512) fp32.
// Memory-bound: ~33.5MB read + ~268MB written once -> ~13us at 23.3 TB/s.
// Strategy: one block per output row (2D grid: x=output row, y=image), async-
// stage the two source rows into LDS (GLOBAL_LOAD_ASYNC_TO_LDS_B64, ASYNCcnt),
// compute with the fixed 0.25/0.75 half-pixel weights, and emit 16B
// nontemporal stores so the 268MB write-once stream does not thrash the
// 192MB L2 (the 33.5MB input lives in L2 and is reused ~4x across rows).

typedef float v4f __attribute__((ext_vector_type(4)));
typedef int   v2i __attribute__((ext_vector_type(2)));

#define AS1 __attribute__((address_space(1)))
#define AS3 __attribute__((address_space(3)))

#define BALANCE 0.5f

#if defined(__AMDGCN__) && __has_builtin(__builtin_amdgcn_global_load_async_to_lds_b64)
#define HAVE_ASYNC_LDS 1
#else
#define HAVE_ASYNC_LDS 0
#endif

#if defined(__AMDGCN__) && __has_builtin(__builtin_amdgcn_s_wait_asynccnt)
#define WAIT_ASYNC() __builtin_amdgcn_s_wait_asynccnt(0)
#else
#define WAIT_ASYNC() asm volatile("s_wait_asynccnt 0" ::: "memory")
#endif

// Compile-time shapes (from the reference): strides become shifts, no scalar
// integer division anywhere.
constexpr int H = 512, W = 512;
constexpr int OH = 2 * H, OW = 2 * W;

__global__ __launch_bounds__(W / 2) void
Interp2xBoundary2d_kernel(const float* __restrict__ x,
                          float* __restrict__ out,
                          float* __restrict__ mask)
{
    __shared__ float lds[2 * W];             // [0,W) top source row, [W,2W) bottom source row
    const int t   = threadIdx.x;             // 0 .. W/2-1 (each thread -> 4 output pixels)
    const int oy  = blockIdx.x;              // output row (no division: 2D grid)
    const int img = blockIdx.y;              // flattened (b,c) image index

    // Vertical taps for 2x half-pixel resize: fixed weights by parity.
    // even oy=2k: rows (k-1, k) weights (0.25, 0.75); odd oy=2k+1: rows (k, k+1) weights (0.75, 0.25)
    const int   k   = oy >> 1;
    const int   odd = oy & 1;
    const int   y0  = odd ? k : (k > 0 ? k - 1 : 0);
    const int   y1  = odd ? (k < H - 1 ? k + 1 : H - 1) : k;
    const float wt  = odd ? 0.75f : 0.25f;   // weight of row y0
    const float wb  = 1.0f - wt;             // weight of row y1

    const int c = t * 2;                     // this thread stages input cols c, c+1
    const float* row0 = x + ((size_t)img * H + y0) * (size_t)W;
    const float* row1 = x + ((size_t)img * H + y1) * (size_t)W;

#if HAVE_ASYNC_LDS
    // CDNA5 async DMA: global -> LDS without touching VGPRs, tracked by ASYNCcnt.
    __builtin_amdgcn_global_load_async_to_lds_b64(
        (AS1 v2i*)(row0 + c),
        (AS3 v2i*)&lds[c], 0, 0);
    __builtin_amdgcn_global_load_async_to_lds_b64(
        (AS1 v2i*)(row1 + c),
        (AS3 v2i*)&lds[W + c], 0, 0);
    WAIT_ASYNC();
    __syncthreads();
#else
    lds[c]         = row0[c];
    lds[c + 1]     = row0[c + 1];
    lds[W + c]     = row1[c];
    lds[W + c + 1] = row1[c + 1];
    __syncthreads();
#endif

    // Output pixels 4t..4t+3 need input cols 2t-1 .. 2t+2 (clamped at the edges,
    // matching the reference's clipped indices; the boundary test uses the
    // clamped duplicates too).
    const int cm1 = (c > 0) ? c - 1 : 0;
    const int cp2 = (c + 2 < W) ? c + 2 : W - 1;

    const float a0 = lds[cm1],     b0 = lds[c],     c0 = lds[c + 1],     d0 = lds[cp2];
    const float a1 = lds[W + cm1], b1 = lds[W + c], c1 = lds[W + c + 1], d1 = lds[W + cp2];

    // Horizontal 2x interpolation (fixed 0.25/0.75 weights).
    const float h0t = 0.25f * a0 + 0.75f * b0;   // ox = 4t   : cols (c-1, c)
    const float h1t = 0.75f * b0 + 0.25f * c0;   // ox = 4t+1 : cols (c, c+1)
    const float h2t = 0.25f * b0 + 0.75f * c0;   // ox = 4t+2 : cols (c, c+1)
    const float h3t = 0.75f * c0 + 0.25f * d0;   // ox = 4t+3 : cols (c+1, c+2)
    const float h0b = 0.25f * a1 + 0.75f * b1;
    const float h1b = 0.75f * b1 + 0.25f * c1;
    const float h2b = 0.25f * b1 + 0.75f * c1;
    const float h3b = 0.75f * c1 + 0.25f * d1;

    v4f o;
    o.x = wt * h0t + wb * h0b;
    o.y = wt * h1t + wb * h1b;
    o.z = wt * h2t + wb * h2b;
    o.w = wt * h3t + wb * h3b;

    // Boundary flag: (min(taps)-0.5)*(max(taps)-0.5) < 0.
    // Pixels 4t+1 and 4t+2 share the exact same 4 taps -> identical mask.
    const float mn0 = fminf(fminf(a0, b0), fminf(a1, b1));
    const float mx0 = fmaxf(fmaxf(a0, b0), fmaxf(a1, b1));
    const float mn1 = fminf(fminf(b0, c0), fminf(b1, c1));
    const float mx1 = fmaxf(fmaxf(b0, c0), fmaxf(b1, c1));
    const float mn3 = fminf(fminf(c0, d0), fminf(c1, d1));
    const float mx3 = fmaxf(fmaxf(c0, d0), fmaxf(c1, d1));

    v4f m;
    m.x = ((mn0 - BALANCE) * (mx0 - BALANCE) < 0.0f) ? 1.0f : 0.0f;
    m.y = ((mn1 - BALANCE) * (mx1 - BALANCE) < 0.0f) ? 1.0f : 0.0f;
    m.z = m.y;
    m.w = ((mn3 - BALANCE) * (mx3 - BALANCE) < 0.0f) ? 1.0f : 0.0f;

    // Coalesced 16B nontemporal stores: write-once stream, keep L2 for input.
    const size_t obase = ((size_t)img * OH + oy) * (size_t)OW + (size_t)(4 * t);
    __builtin_nontemporal_store(o, (v4f*)(out + obase));
    __builtin_nontemporal_store(m, (v4f*)(mask + obase));
}

extern "C" void kernel_launch(void* const* d_in, const int* in_sizes, int n_in,
                              void* d_out, int out_size, void* d_ws, size_t ws_size,
                              hipStream_t stream) {
    (void)in_sizes; (void)n_in; (void)d_ws; (void)ws_size; (void)out_size;

    // Reference shapes: x is (B,C,H,W) = (4,8,512,512) fp32.
    const int B = 4, C = 8;

    const float* x = (const float*)d_in[0];
    float* out     = (float*)d_out;                        // first output (fp32)
    float* mask    = out + (size_t)B * C * OH * OW;        // second output (bool as 0/1 float)

    dim3 block(W / 2);        // 256 threads = 8 wave32s; each thread -> 4 output pixels (full row)
    dim3 grid(OH, B * C);     // x = output row (1024), y = image (32); no integer division in-kernel

    Interp2xBoundary2d_kernel<<<grid, block, 0, stream>>>(x, out, mask);
}